// OpensceneEncoder_27797028340106
// MI455X (gfx1250) — compile-verified
//
#include <hip/hip_runtime.h>
#include <hip/hip_bf16.h>

#define NPTS   40000
#define NGROUP 256
#define KNN    64
#define FDIM   768
#define FLT_BIG 3.402823e38f

typedef __attribute__((ext_vector_type(2))) float v2f;
typedef __attribute__((ext_vector_type(8))) float v8f;

// ---------------------------------------------------------------------------
// Kernel A: farthest-point sampling. One 1024-thread block per batch.
// dist[] (40000 f32 = 160KB) lives in LDS for all 255 serial iterations;
// xyz (480KB/batch) stays hot in L2.
// ---------------------------------------------------------------------------
__global__ __launch_bounds__(1024) void fps_kernel(const float* __restrict__ xyz,
                                                   float* __restrict__ centers_ws,
                                                   float* __restrict__ out_centers,
                                                   int N)
{
    extern __shared__ float smem[];
    float* dist = smem;              // N floats
    float* rval = smem + N;          // 32 floats (per-wave partials)
    int*   ridx = (int*)(rval + 32); // 32 ints
    __shared__ int s_last;

    const int b   = blockIdx.x;
    const int tid = threadIdx.x;
    const float* X = xyz + (size_t)b * N * 3;
    float* cw = centers_ws + (size_t)b * NGROUP * 3;
    float* co = out_centers + (size_t)b * NGROUP * 3;

    for (int i = tid; i < N; i += 1024) dist[i] = 1e10f;
    if (tid == 0) {
        s_last = 0;
        cw[0] = co[0] = X[0];
        cw[1] = co[1] = X[1];
        cw[2] = co[2] = X[2];
    }
    __syncthreads();

    for (int s = 1; s < NGROUP; ++s) {
        const int last = s_last;
        const float px = X[(size_t)last * 3 + 0];
        const float py = X[(size_t)last * 3 + 1];
        const float pz = X[(size_t)last * 3 + 2];

        float best = -1.0f;
        int   bi   = 0x7fffffff;
        for (int i = tid; i < N; i += 1024) {
            float dx = X[(size_t)i * 3 + 0] - px;
            float dy = X[(size_t)i * 3 + 1] - py;
            float dz = X[(size_t)i * 3 + 2] - pz;
            float d  = dx * dx + dy * dy + dz * dz;
            float od = dist[i];
            d = fminf(od, d);
            dist[i] = d;
            if (d > best || (d == best && i < bi)) { best = d; bi = i; }
        }
        // wave32 argmax reduce (prefer lower index on ties, matching jnp.argmax)
        #pragma unroll
        for (int off = 16; off > 0; off >>= 1) {
            float ov = __shfl_xor(best, off, 32);
            int   oi = __shfl_xor(bi,   off, 32);
            if (ov > best || (ov == best && oi < bi)) { best = ov; bi = oi; }
        }
        if ((tid & 31) == 0) { rval[tid >> 5] = best; ridx[tid >> 5] = bi; }
        __syncthreads();
        if (tid < 32) {
            best = rval[tid];
            bi   = ridx[tid];
            #pragma unroll
            for (int off = 16; off > 0; off >>= 1) {
                float ov = __shfl_xor(best, off, 32);
                int   oi = __shfl_xor(bi,   off, 32);
                if (ov > best || (ov == best && oi < bi)) { best = ov; bi = oi; }
            }
            if (tid == 0) {
                s_last = bi;
                cw[(size_t)s * 3 + 0] = co[(size_t)s * 3 + 0] = X[(size_t)bi * 3 + 0];
                cw[(size_t)s * 3 + 1] = co[(size_t)s * 3 + 1] = X[(size_t)bi * 3 + 1];
                cw[(size_t)s * 3 + 2] = co[(size_t)s * 3 + 2] = X[(size_t)bi * 3 + 2];
            }
        }
        __syncthreads();
    }
}

// ---------------------------------------------------------------------------
// Kernel B: rank-equivalent distance matrix via V_WMMA_F32_16X16X4_F32.
//   d'[m][n] = |x_n|^2 - 2 c_m . x_n    (|c_m|^2 dropped: constant per row)
// A row (K=4)    = (-2cx, -2cy, -2cz, 1)
// B column (K=4) = ( x,    y,    z,   |x|^2)
// Grid: (B*16 center-tiles) x (NSPLIT point slices) for machine fill.
// Each wave produces a full 16x16 D tile per WMMA op.
// ---------------------------------------------------------------------------
#define NSPLIT 25   // 40000 / 25 = 1600 points (100 tiles) per slice

__global__ __launch_bounds__(256) void knn_dist_kernel(const float* __restrict__ xyz,
                                                       const float* __restrict__ centers,
                                                       float* __restrict__ distws,
                                                       int N)
{
    const int bidx  = blockIdx.x;          // b * 16 + mtile
    const int b     = bidx >> 4;
    const int mtile = bidx & 15;
    const int lane  = threadIdx.x & 31;
    const int wave  = threadIdx.x >> 5;    // 0..7
    const int half  = (lane < 16) ? 0 : 1;
    const int l16   = lane & 15;

    const int slice = N / NSPLIT;          // 1600, multiple of 16
    const int nbeg  = blockIdx.y * slice;
    const int nend  = nbeg + slice;

    const float* X = xyz + (size_t)b * N * 3;
    const float* C = centers + (size_t)(b * NGROUP + mtile * 16) * 3;
    float* drow = distws + (size_t)(b * NGROUP + mtile * 16) * N;

    // A operand: lane l16 = center row M; half 0 carries K=0,1 ; half 1 K=2,3
    const float cx = C[(size_t)l16 * 3 + 0];
    const float cy = C[(size_t)l16 * 3 + 1];
    const float cz = C[(size_t)l16 * 3 + 2];
    v2f a;
    a.x = half ? (-2.0f * cz) : (-2.0f * cx);
    a.y = half ? 1.0f         : (-2.0f * cy);

    for (int n0 = nbeg + wave * 16; n0 < nend; n0 += 16 * 8) {
        const int n = n0 + l16;
        const float px = X[(size_t)n * 3 + 0];
        const float py = X[(size_t)n * 3 + 1];
        const float pz = X[(size_t)n * 3 + 2];
        const float n2 = px * px + py * py + pz * pz;
        v2f bb;
        bb.x = half ? pz : px;
        bb.y = half ? n2 : py;

        v8f c = {};
        c = __builtin_amdgcn_wmma_f32_16x16x4_f32(
                /*neg_a=*/false, a, /*neg_b=*/false, bb,
                /*c_mod=*/(short)0, c, /*reuse_a=*/false, /*reuse_b=*/false);

        // D tile: VGPR r -> M = r + 8*half, N = n (lanes 0..15 consecutive n)
        #pragma unroll
        for (int r = 0; r < 8; ++r) {
            const int mm = r + 8 * half;
            drow[(size_t)mm * N + n] = c[r];
        }
    }
}

// ---------------------------------------------------------------------------
// Kernel C: per (batch, center): stream the distance row into LDS, then
// tournament top-64: each thread keeps the argmin "head" of its strided
// 40-element slice in registers; each round is a shuffle+LDS reduce over the
// 1024 heads, and only the winning thread rescans its slice. Then coalesced
// gather-mean of the 768-dim feature rows + validity mask.
// ---------------------------------------------------------------------------
__global__ __launch_bounds__(1024) void select_agg_kernel(const float* __restrict__ distws,
                                                          const float* __restrict__ feats,
                                                          const int*   __restrict__ valid,
                                                          float* __restrict__ out_fts,
                                                          float* __restrict__ out_mask,
                                                          int N)
{
    extern __shared__ float smem[];
    float* dist = smem;              // N floats
    float* rval = smem + N;          // 32
    int*   ridx = (int*)(rval + 32); // 32
    __shared__ int sel[KNN];
    __shared__ int s_win;

    const int gc  = blockIdx.x;      // b * NGROUP + g
    const int b   = gc >> 8;
    const int tid = threadIdx.x;

    const float* drow = distws + (size_t)gc * N;
    for (int i = tid; i < N; i += 1024) dist[i] = drow[i];
    __syncthreads();

    // Per-thread head = argmin over the thread's strided slice.
    float hv = FLT_BIG;
    int   hi = 0x7fffffff;
    for (int i = tid; i < N; i += 1024) {
        float d = dist[i];
        if (d < hv || (d == hv && i < hi)) { hv = d; hi = i; }
    }

    for (int k = 0; k < KNN; ++k) {
        float best = hv;
        int   bi   = hi;
        #pragma unroll
        for (int off = 16; off > 0; off >>= 1) {
            float ov = __shfl_xor(best, off, 32);
            int   oi = __shfl_xor(bi,   off, 32);
            if (ov < best || (ov == best && oi < bi)) { best = ov; bi = oi; }
        }
        if ((tid & 31) == 0) { rval[tid >> 5] = best; ridx[tid >> 5] = bi; }
        __syncthreads();
        if (tid < 32) {
            best = rval[tid];
            bi   = ridx[tid];
            #pragma unroll
            for (int off = 16; off > 0; off >>= 1) {
                float ov = __shfl_xor(best, off, 32);
                int   oi = __shfl_xor(bi,   off, 32);
                if (ov < best || (ov == best && oi < bi)) { best = ov; bi = oi; }
            }
            if (tid == 0) { sel[k] = bi; s_win = bi; }
        }
        __syncthreads();
        const int win = s_win;
        if ((win & 1023) == tid) {     // owner of the winning element
            dist[win] = FLT_BIG;
            hv = FLT_BIG; hi = 0x7fffffff;
            for (int i = tid; i < N; i += 1024) {
                float d = dist[i];
                if (d < hv || (d == hv && i < hi)) { hv = d; hi = i; }
            }
        }
    }
    __syncthreads();

    // Mean of the 768-dim feature rows of the 64 selected points.
    const float* F = feats + (size_t)b * N * FDIM;
    if (tid < FDIM) {
        float acc = 0.0f;
        #pragma unroll 4
        for (int k = 0; k < KNN; ++k) {
            const float* row = F + (size_t)sel[k] * FDIM;
            if (k + 1 < KNN) __builtin_prefetch(F + (size_t)sel[k + 1] * FDIM + tid, 0, 0);
            acc += row[tid];
        }
        out_fts[(size_t)gc * FDIM + tid] = acc * (1.0f / 64.0f);
    }
    if (tid == 0) {
        const int* V = valid + (size_t)b * N;
        int sv = 0;
        #pragma unroll 4
        for (int k = 0; k < KNN; ++k) sv += V[sel[k]];
        out_mask[gc] = (sv > 0) ? 1.0f : 0.0f;
    }
}

// ---------------------------------------------------------------------------
extern "C" void kernel_launch(void* const* d_in, const int* in_sizes, int n_in,
                              void* d_out, int out_size, void* d_ws, size_t ws_size,
                              hipStream_t stream) {
    (void)n_in; (void)out_size; (void)ws_size;
    const float* xyzs  = (const float*)d_in[0];
    const float* feats = (const float*)d_in[1];
    const int*   valid = (const int*)d_in[2];

    const int B = 2;
    const int N = in_sizes[0] / (B * 3);   // 40000

    // Output layout: all_fts [B*256*768] ++ mask [B*256] ++ scene_center [B*256*3]
    float* out_fts  = (float*)d_out;
    float* out_mask = out_fts + (size_t)B * NGROUP * FDIM;
    float* out_cent = out_mask + (size_t)B * NGROUP;

    // Workspace: centers [B*256*3] then distance matrix [B*256*N] (82MB, fits L2)
    float* centers_ws = (float*)d_ws;
    float* distws     = centers_ws + (size_t)B * NGROUP * 3;

    const size_t smem_bytes = (size_t)N * sizeof(float) + 32 * sizeof(float) + 32 * sizeof(int);

    fps_kernel<<<B, 1024, smem_bytes, stream>>>(xyzs, centers_ws, out_cent, N);
    knn_dist_kernel<<<dim3(B * (NGROUP / 16), NSPLIT), 256, 0, stream>>>(xyzs, centers_ws,
                                                                         distws, N);
    select_agg_kernel<<<B * NGROUP, 1024, smem_bytes, stream>>>(distws, feats, valid,
                                                                out_fts, out_mask, N);
}